// sparse_attn_52578989638161
// MI455X (gfx1250) — compile-verified
//
#include <hip/hip_runtime.h>
#include <hip/hip_bf16.h>

#define D 128
#define TILE 16

typedef __attribute__((ext_vector_type(2))) float v2f;
typedef __attribute__((ext_vector_type(8))) float v8f;

// D = A(16x4, f32) * B(4x16, f32) + C(16x16, f32)
__device__ __forceinline__ v8f wmma_f32_k4(v2f a, v2f b, v8f c) {
    // (neg_a, A, neg_b, B, c_mod, C, reuse_a, reuse_b)
    return __builtin_amdgcn_wmma_f32_16x16x4_f32(
        false, a, false, b, (short)0, c, false, false);
}

// One wave computes a 16x16 tile of OUT = A @ W^T + bias, K = 128.
// A: [*, 128] row-major (rows i0..i0+15), W: [128,128] row-major.
// ISA operand layouts (CDNA5 ISA §7.12.2):
//   A 16x4 : lane l -> M = l&15 ; VGPR j -> K = (l>>4)*2 + j
//   B 4x16 : lane l -> N = l&15 ; VGPR j -> K = (l>>4)*2 + j   (B[k][n] = W[n][k])
//   C 16x16: lane l, VGPR r -> M = r + 8*(l>>4), N = l&15
__device__ __forceinline__ void gemm_tile_wave(
    const float* __restrict__ A, const float* __restrict__ W,
    const float* __restrict__ bias, float* __restrict__ OUT,
    int i0, int jt, int lane)
{
    const int lane16 = lane & 15;
    const int half   = lane >> 4;

    const float bval = bias[jt * TILE + lane16];
    v8f c;
#pragma unroll
    for (int r = 0; r < 8; ++r) c[r] = bval;

    const float* arow = A + (size_t)(i0 + lane16) * D + half * 2;
    const float* brow = W + (size_t)(jt * TILE + lane16) * D + half * 2;

#pragma unroll
    for (int kb = 0; kb < D; kb += 4) {
        v2f a = *(const v2f*)(arow + kb);
        v2f b = *(const v2f*)(brow + kb);
        c = wmma_f32_k4(a, b, c);
    }

#pragma unroll
    for (int r = 0; r < 8; ++r) {
        OUT[(size_t)(i0 + r + 8 * half) * D + jt * TILE + lane16] = c[r];
    }
}

// K1: three fused GEMMs sharing A = x. 256 threads = 8 waves; wave w owns
// column tile jt = w for all three matrices. Grid.x = N/16.
__global__ __launch_bounds__(256) void k_gemm3(
    const float* __restrict__ x,
    const float* __restrict__ Ws, const float* __restrict__ bs,
    const float* __restrict__ Ww, const float* __restrict__ bw,
    const float* __restrict__ Wv, const float* __restrict__ bv,
    float* __restrict__ lraw, float* __restrict__ wraw, float* __restrict__ v)
{
    const int i0   = blockIdx.x * TILE;
    const int wave = threadIdx.x >> 5;
    const int lane = threadIdx.x & 31;
    gemm_tile_wave(x, Ws, bs, lraw, i0, wave, lane);
    gemm_tile_wave(x, Ww, bw, wraw, i0, wave, lane);
    gemm_tile_wave(x, Wv, bv, v,    i0, wave, lane);
}

// K2: per-row softmax of logits -> truncated int index; softmax of wlog -> w
// (in place). 128 threads per row, grid = N.
__global__ __launch_bounds__(128) void k_softmax_idx(
    const float* __restrict__ lraw, float* __restrict__ wbuf,
    int* __restrict__ idx, int N)
{
    __shared__ float red[D];
    const int i = blockIdx.x;
    const int t = threadIdx.x;

    // ---- sampling logits -> idx ----
    float lv = lraw[(size_t)i * D + t];
    red[t] = lv; __syncthreads();
#pragma unroll
    for (int s = 64; s > 0; s >>= 1) {
        if (t < s) red[t] = fmaxf(red[t], red[t + s]);
        __syncthreads();
    }
    float m = red[0]; __syncthreads();
    float e = expf(lv - m);
    red[t] = e; __syncthreads();
#pragma unroll
    for (int s = 64; s > 0; s >>= 1) {
        if (t < s) red[t] += red[t + s];
        __syncthreads();
    }
    float p = e / red[0]; __syncthreads();
    idx[(size_t)i * D + t] = (int)(p * (float)N);   // truncation, matches astype(int32)

    // ---- sampling weights ----
    float wv = wbuf[(size_t)i * D + t];
    red[t] = wv; __syncthreads();
#pragma unroll
    for (int s = 64; s > 0; s >>= 1) {
        if (t < s) red[t] = fmaxf(red[t], red[t + s]);
        __syncthreads();
    }
    float mw = red[0]; __syncthreads();
    float ew = expf(wv - mw);
    red[t] = ew; __syncthreads();
#pragma unroll
    for (int s = 64; s > 0; s >>= 1) {
        if (t < s) red[t] += red[t + s];
        __syncthreads();
    }
    wbuf[(size_t)i * D + t] = ew / red[0];
}

// K3: rows[i,t] = sum_k w[i,k] * v[idx[i,t],k]. w row staged in LDS; v reads
// hit L2 (v table is 8 MB, L2 is 192 MB). 128 threads per row, grid = N.
__global__ __launch_bounds__(128) void k_gather_dot(
    const float* __restrict__ w, const int* __restrict__ idx,
    const float* __restrict__ v, float* __restrict__ rows)
{
    __shared__ float wsh[D];
    const int i = blockIdx.x;
    const int t = threadIdx.x;
    wsh[t] = w[(size_t)i * D + t];
    __syncthreads();

    const int j = idx[(size_t)i * D + t];
    const float4* vr = (const float4*)(v + (size_t)j * D);
    float acc = 0.0f;
#pragma unroll
    for (int k = 0; k < D / 4; ++k) {
        float4 q = vr[k];
        acc += wsh[4 * k + 0] * q.x + wsh[4 * k + 1] * q.y
             + wsh[4 * k + 2] * q.z + wsh[4 * k + 3] * q.w;
    }
    rows[(size_t)i * D + t] = acc;
}

// K4: out = rows @ Wo^T + bo. 256 threads = 8 waves, one column tile each.
__global__ __launch_bounds__(256) void k_gemm_out(
    const float* __restrict__ rows,
    const float* __restrict__ Wo, const float* __restrict__ bo,
    float* __restrict__ out)
{
    const int i0   = blockIdx.x * TILE;
    const int wave = threadIdx.x >> 5;
    const int lane = threadIdx.x & 31;
    gemm_tile_wave(rows, Wo, bo, out, i0, wave, lane);
}

extern "C" void kernel_launch(void* const* d_in, const int* in_sizes, int n_in,
                              void* d_out, int out_size, void* d_ws, size_t ws_size,
                              hipStream_t stream) {
    const float* x  = (const float*)d_in[0];
    const float* Ws = (const float*)d_in[1];
    const float* bs = (const float*)d_in[2];
    const float* Ww = (const float*)d_in[3];
    const float* bw = (const float*)d_in[4];
    const float* Wv = (const float*)d_in[5];
    const float* bv = (const float*)d_in[6];
    const float* Wo = (const float*)d_in[7];
    const float* bo = (const float*)d_in[8];
    float* out = (float*)d_out;

    const int N = in_sizes[0] / D;           // 16384
    const size_t rowElems = (size_t)N * D;   // 2M floats = 8 MB

    // workspace layout (32 MB total): lraw | w | v | idx ; rows reuses lraw
    float* lraw = (float*)d_ws;
    float* wbuf = lraw + rowElems;
    float* vbuf = wbuf + rowElems;
    int*   idxb = (int*)(vbuf + rowElems);
    float* rows = lraw;                      // lraw dead after K2

    const int rowTiles = N / TILE;           // 1024

    k_gemm3<<<rowTiles, 256, 0, stream>>>(x, Ws, bs, Ww, bw, Wv, bv,
                                          lraw, wbuf, vbuf);
    k_softmax_idx<<<N, D, 0, stream>>>(lraw, wbuf, idxb, N);
    k_gather_dot<<<N, D, 0, stream>>>(wbuf, idxb, vbuf, rows);
    k_gemm_out<<<rowTiles, 256, 0, stream>>>(rows, Wo, bo, out);
}